// Quantization_14774687498745
// MI455X (gfx1250) — compile-verified
//
#include <hip/hip_runtime.h>
#include <hip/hip_bf16.h>

typedef __attribute__((ext_vector_type(2))) float v2f;
typedef __attribute__((ext_vector_type(8))) float v8f;
typedef __attribute__((ext_vector_type(4))) unsigned int v4u;
typedef __attribute__((ext_vector_type(4))) int v4i;
typedef __attribute__((ext_vector_type(8))) int v8i;

#define N_ROWS 65536
#define DIM    256
#define K_CODES 4096
#define NC     16      // codebook codes per LDS chunk
#define DP     260     // padded LDS row stride (dwords): 260%64=4 -> conflict-free B reads
#define WAVES  8
#define ROWS_PER_WAVE 32
#define ROWS_PER_WG   (WAVES * ROWS_PER_WAVE)   // 256

#if defined(__has_builtin)
#  if __has_builtin(__builtin_amdgcn_tensor_load_to_lds)
#    define USE_TDM 1
#  endif
#endif

// ---------------------------------------------------------------------------
// Pre-pass: cnorm[k] = ||codebook[k]||^2  (one wave per code)
// ---------------------------------------------------------------------------
__global__ void __launch_bounds__(256) vq_cnorm(const float* __restrict__ cb,
                                                float* __restrict__ cnorm) {
    const int lane = threadIdx.x & 31;
    const int wave = threadIdx.x >> 5;
    const int code = blockIdx.x * 8 + wave;
    const float4* cp = (const float4*)(cb + (size_t)code * DIM);
    float4 a = cp[lane];
    float4 b = cp[lane + 32];
    float s = a.x*a.x + a.y*a.y + a.z*a.z + a.w*a.w
            + b.x*b.x + b.y*b.y + b.z*b.z + b.w*b.w;
#pragma unroll
    for (int off = 16; off > 0; off >>= 1) s += __shfl_xor(s, off);
    if (lane == 0) cnorm[code] = s;
}

#ifdef USE_TDM
// Issue one TDM descriptor: DMA a 16x256 f32 codebook tile into LDS with
// 4-dword padding after every 256 dwords (row stride 260 dw = our layout).
__device__ __forceinline__ void tdm_stage_chunk(const float* gsrc,
                                                unsigned ldsByteAddr) {
    unsigned long long ga = (unsigned long long)(size_t)gsrc;
    v4u g0;
    g0[0] = 1u;                                       // count=1, user descriptor
    g0[1] = ldsByteAddr;                              // lds_addr (bytes)
    g0[2] = (unsigned)(ga & 0xFFFFFFFFu);             // global_addr[31:0]
    g0[3] = (unsigned)((ga >> 32) & 0x01FFFFFFu)      // global_addr[56:32]
          | (2u << 30);                               // type=2 ("image")
    v8i g1;
    g1[0] = (int)((2u << 16)      // data_size = 4 bytes
                | (1u << 20)      // pad_enable
                | (7u << 22)      // pad_interval: 256 dwords
                | (3u << 25));    // pad_amount:   4 dwords
    g1[1] = (int)(256u << 16);    // [47:32] atomic_barrier=0 | tensor_dim0=256 (lo)
    g1[2] = (int)(4096u << 16);   // tensor_dim0 hi=0 | tensor_dim1=4096 (lo)
    g1[3] = (int)(256u << 16);    // tensor_dim1 hi=0 | tile_dim0=256
    g1[4] = 16;                   // tile_dim1=16 | tile_dim2=0
    g1[5] = 256;                  // tensor_dim0_stride = 256 (lo)
    g1[6] = 0;                    // stride0 hi | tensor_dim1_stride lo
    g1[7] = 0;
    v4i z4 = {0, 0, 0, 0};
#if __clang_major__ >= 23
    v8i z8 = {0, 0, 0, 0, 0, 0, 0, 0};
    __builtin_amdgcn_tensor_load_to_lds(g0, g1, z4, z4, z8, 0);
#else
    __builtin_amdgcn_tensor_load_to_lds(g0, g1, z4, z4, 0);
#endif
}

__device__ __forceinline__ void tdm_wait() {
#if __has_builtin(__builtin_amdgcn_s_wait_tensorcnt)
    __builtin_amdgcn_s_wait_tensorcnt(0);
#else
    asm volatile("s_wait_tensorcnt 0x0" ::: "memory");
#endif
}
#endif  // USE_TDM

// ---------------------------------------------------------------------------
// Main: WMMA f32 GEMM (score = cnorm - 2*z.c^T) + argmin + gather
// ---------------------------------------------------------------------------
__global__ void __launch_bounds__(256) vq_main(const float* __restrict__ z,
                                               const float* __restrict__ cb,
                                               const float* __restrict__ cnorm,
                                               float* __restrict__ zq,
                                               float* __restrict__ idsOut,
                                               float* __restrict__ lossAcc) {
    extern __shared__ float smem[];   // 2 buffers * NC * DP floats
    const int lane = threadIdx.x & 31;
    const int lm   = lane & 15;       // column / row-within-tile lane id
    const int h    = lane >> 4;       // lane half (K sub-slice select)
    const int wave = threadIdx.x >> 5;
    const int waveRow = blockIdx.x * ROWS_PER_WG + wave * ROWS_PER_WAVE;

    // ---- Load two 16x256 A tiles of z into registers in WMMA A layout ----
    v2f a[2][64];
#pragma unroll
    for (int t = 0; t < 2; ++t) {
        const v2f* zp = (const v2f*)(z + (size_t)(waveRow + t * 16 + lm) * DIM);
#pragma unroll
        for (int kk = 0; kk < 64; ++kk)
            a[t][kk] = zp[2 * kk + h];
    }

    float    best[2][8];
    unsigned bi[2][8];
#pragma unroll
    for (int t = 0; t < 2; ++t)
#pragma unroll
        for (int g = 0; g < 8; ++g) { best[t][g] = 3.4e38f; bi[t][g] = 0u; }

    const unsigned smemBase = (unsigned)(size_t)(void*)smem;

    // ---- stage chunk 0 into buffer 0 ----
#ifdef USE_TDM
    if (threadIdx.x < 32) {
        tdm_stage_chunk(cb, smemBase);
        tdm_wait();
    }
#else
    {
        const int cc = threadIdx.x >> 4, seg = threadIdx.x & 15;
        const float4* src = (const float4*)(cb + (size_t)cc * DIM + seg * 16);
        float4* dst = (float4*)(smem + cc * DP + seg * 16);
#pragma unroll
        for (int i = 0; i < 4; ++i) dst[i] = src[i];
    }
#endif
    __syncthreads();

    const int NCHUNK = K_CODES / NC;  // 256
    for (int ch = 0; ch < NCHUNK; ++ch) {
        const int buf = (ch & 1) * (NC * DP);

        // ---- async-prefetch next chunk into the other buffer ----
        if (ch + 1 < NCHUNK) {
            const int nb = ((ch + 1) & 1) * (NC * DP);
#ifdef USE_TDM
            if (threadIdx.x < 32)
                tdm_stage_chunk(cb + (size_t)(ch + 1) * NC * DIM,
                                smemBase + (unsigned)nb * 4u);
#else
            const int cc = threadIdx.x >> 4, seg = threadIdx.x & 15;
            const float4* src =
                (const float4*)(cb + (size_t)((ch + 1) * NC + cc) * DIM + seg * 16);
            float4* dst = (float4*)(smem + nb + cc * DP + seg * 16);
#pragma unroll
            for (int i = 0; i < 4; ++i) dst[i] = src[i];
#endif
        }

        // ---- GEMM: two 16x16 C tiles += A * B over K=256 in steps of 4 ----
        v8f c0 = {0.f, 0.f, 0.f, 0.f, 0.f, 0.f, 0.f, 0.f};
        v8f c1 = {0.f, 0.f, 0.f, 0.f, 0.f, 0.f, 0.f, 0.f};
        const v2f* bp = (const v2f*)(smem + buf + lm * DP);
#pragma unroll
        for (int kk = 0; kk < 64; ++kk) {
            v2f b = bp[2 * kk + h];
            c0 = __builtin_amdgcn_wmma_f32_16x16x4_f32(
                     false, a[0][kk], false, b, (short)0, c0, false, false);
            c1 = __builtin_amdgcn_wmma_f32_16x16x4_f32(
                     false, a[1][kk], false, b, (short)0, c1, false, false);
        }

        // ---- running argmin update: score = cnorm[n] - 2*dot ----
        const float    cn   = cnorm[ch * NC + lm];
        const unsigned nIdx = (unsigned)(ch * NC + lm);
#pragma unroll
        for (int g = 0; g < 8; ++g) {
            float s0 = cn - 2.0f * c0[g];
            if (s0 < best[0][g]) { best[0][g] = s0; bi[0][g] = nIdx; }
            float s1 = cn - 2.0f * c1[g];
            if (s1 < best[1][g]) { best[1][g] = s1; bi[1][g] = nIdx; }
        }

#ifdef USE_TDM
        if (threadIdx.x < 32) tdm_wait();
#endif
        __syncthreads();
    }

    // ---- epilogue: cross-lane argmin, id writes, gather z_q, loss ----
    float lsum = 0.0f;
#pragma unroll
    for (int t = 0; t < 2; ++t) {
#pragma unroll
        for (int g = 0; g < 8; ++g) {
            float    d = best[t][g];
            unsigned i = bi[t][g];
#pragma unroll
            for (int off = 8; off > 0; off >>= 1) {  // stays within 16-lane half
                float    od = __shfl_xor(d, off);
                unsigned oi = __shfl_xor(i, off);
                if (od < d || (od == d && oi < i)) { d = od; i = oi; }
            }
            const unsigned id0 = (unsigned)__shfl((int)i, 0);   // half 0: row g
            const unsigned id1 = (unsigned)__shfl((int)i, 16);  // half 1: row g+8
#pragma unroll
            for (int hh = 0; hh < 2; ++hh) {
                const unsigned id  = hh ? id1 : id0;
                const int      row = waveRow + t * 16 + g + 8 * hh;
                if (lane == 0) idsOut[row] = (float)id;
                const float4* cr = (const float4*)(cb + (size_t)id * DIM);
                const float4* zr = (const float4*)(z + (size_t)row * DIM);
                float4*       qr = (float4*)(zq + (size_t)row * DIM);
#pragma unroll
                for (int j = 0; j < 2; ++j) {
                    const int idx = lane + 32 * j;
                    float4 cv = cr[idx];
                    float4 zv = zr[idx];
                    qr[idx] = cv;
                    float dx = cv.x - zv.x, dy = cv.y - zv.y;
                    float dz = cv.z - zv.z, dw = cv.w - zv.w;
                    lsum += dx * dx + dy * dy + dz * dz + dw * dw;
                }
            }
        }
    }
#pragma unroll
    for (int off = 16; off > 0; off >>= 1) lsum += __shfl_xor(lsum, off);
    if (lane == 0) atomicAdd(lossAcc, lsum);
}

// ---------------------------------------------------------------------------
__global__ void vq_finalize(const float* __restrict__ lossAcc,
                            float* __restrict__ out) {
    out[0] = lossAcc[0] * 1.25f / 16777216.0f;  // (1+beta) * sum / (N*D)
}

// ---------------------------------------------------------------------------
extern "C" void kernel_launch(void* const* d_in, const int* in_sizes, int n_in,
                              void* d_out, int out_size, void* d_ws, size_t ws_size,
                              hipStream_t stream) {
    (void)in_sizes; (void)n_in; (void)out_size; (void)ws_size;
    const float* z  = (const float*)d_in[0];   // [65536, 1, 256] f32
    const float* cb = (const float*)d_in[1];   // [4096, 256] f32
    float* out     = (float*)d_out;            // zq | loss | ids (flat)
    float* cnorm   = (float*)d_ws;             // [4096]
    float* lossAcc = cnorm + K_CODES;          // [1]

    hipMemsetAsync(lossAcc, 0, sizeof(float), stream);

    vq_cnorm<<<K_CODES / 8, 256, 0, stream>>>(cb, cnorm);

    const size_t smemBytes = 2u * NC * DP * sizeof(float);  // 33,280 B
    float* idsOut = out + (size_t)N_ROWS * DIM + 1;
    vq_main<<<N_ROWS / ROWS_PER_WG, 256, smemBytes, stream>>>(
        z, cb, cnorm, out, idsOut, lossAcc);

    vq_finalize<<<1, 1, 0, stream>>>(lossAcc, out + (size_t)N_ROWS * DIM);
}